// DirectRegressionLoss_54657753808993
// MI455X (gfx1250) — compile-verified
//
#include <hip/hip_runtime.h>
#include <hip/hip_bf16.h>

typedef float v2f __attribute__((ext_vector_type(2)));
typedef float v8f __attribute__((ext_vector_type(8)));

// Problem constants
static constexpr int   kD       = 144;
static constexpr int   kPlane   = kD * kD;          // 20736
static constexpr int   kVol     = kD * kD * kD;     // 2985984
static constexpr int   kN       = 2 * kVol;         // 5971968 voxels total
static constexpr int   kLines   = 2 * kD * kD;      // 41472 W-lines (also # columns for D pass)
static constexpr float kInvWin3 = 1.0f / 1331.0f;   // 1/11^3
static constexpr float kC1      = 0.01f * 0.01f;
static constexpr float kC2      = 0.03f * 0.03f;

__device__ __forceinline__ int iclamp(int v, int lo, int hi) {
    return v < lo ? lo : (v > hi ? hi : v);
}
__device__ __forceinline__ int iabs_(int v) { return v < 0 ? -v : v; }

// ---------------------------------------------------------------------------
// Pass 1: box-sum along W (contiguous axis). One 144-float line per block.
// Builds the 5 fields (p, t, p^2, t^2, p*t) in LDS, writes unnormalized
// 11-tap sums to S1, and emits a deterministic per-block L1 partial.
// ---------------------------------------------------------------------------
__global__ __launch_bounds__(256) void pass1_w(const float* __restrict__ pred,
                                               const float* __restrict__ target,
                                               float* __restrict__ S1,
                                               float* __restrict__ l1_partial) {
    __shared__ float sh[5][kD];
    __shared__ float red[256];

    const int line = blockIdx.x;          // 0 .. kLines-1
    const int base = line * kD;
    const int tid  = threadIdx.x;

    float l1 = 0.0f;
    if (tid < kD) {
        const float p = pred[base + tid];
        const float t = target[base + tid];
        sh[0][tid] = p;
        sh[1][tid] = t;
        sh[2][tid] = p * p;
        sh[3][tid] = t * t;
        sh[4][tid] = p * t;
        l1 = fabsf(p - t);
    }
    red[tid] = l1;
    __syncthreads();

    if (tid < kD) {
        const int lo = tid - 5 < 0 ? 0 : tid - 5;
        const int hi = tid + 5 > kD - 1 ? kD - 1 : tid + 5;
        float s0 = 0.f, s1 = 0.f, s2 = 0.f, s3 = 0.f, s4 = 0.f;
        for (int w = lo; w <= hi; ++w) {
            s0 += sh[0][w]; s1 += sh[1][w]; s2 += sh[2][w];
            s3 += sh[3][w]; s4 += sh[4][w];
        }
        const int o = base + tid;
        S1[0 * kN + o] = s0;
        S1[1 * kN + o] = s1;
        S1[2 * kN + o] = s2;
        S1[3 * kN + o] = s3;
        S1[4 * kN + o] = s4;
    }

    // Deterministic block tree-reduce of |p-t|
    for (int s = 128; s > 0; s >>= 1) {
        if (tid < s) red[tid] += red[tid + s];
        __syncthreads();
    }
    if (tid == 0) l1_partial[line] = red[0];
}

// ---------------------------------------------------------------------------
// Pass 2: box-sum along H as a banded f32 GEMM via V_WMMA_F32_16X16X4_F32.
// Out[h][w] = sum_k Band[h][k] * In[k][w], Band[h][k] = (|h-k|<=5 && k in range).
// One wave per 16x16 output tile of one (field, b, d) plane.
//   A (16x4, MxK): lanes 0-15 hold M=0..15 with K={k0,k0+1}; lanes 16-31 K={k0+2,k0+3}
//   B (4x16, KxN): lanes 0-15 hold N=0..15 with K={k0,k0+1}; lanes 16-31 K={k0+2,k0+3}
//   C/D (16x16):   VGPR v: lanes 0-15 -> M=v, lanes 16-31 -> M=v+8
// ---------------------------------------------------------------------------
__global__ __launch_bounds__(32) void pass2_h(const float* __restrict__ S1,
                                              float* __restrict__ S2) {
    const int lane  = threadIdx.x;        // 0..31, EXEC all ones (no divergence)
    const int n     = lane & 15;
    const int khalf = lane >> 4;

    const int tile = blockIdx.x;          // 0..80
    const int tw   = tile % 9;
    const int th   = tile / 9;
    const int bd   = blockIdx.y;          // 0..287
    const int f    = blockIdx.z;          // 0..4

    const float* __restrict__ in  = S1 + f * kN + bd * kPlane;
    float* __restrict__       out = S2 + f * kN + bd * kPlane;

    const int h0 = th * 16;
    const int w0 = tw * 16;
    const int hm = h0 + n;                // output row this lane contributes A for

    v8f acc = {};

#pragma unroll
    for (int kb = 0; kb < 8; ++kb) {
        const int k0 = h0 - 8 + kb * 4;   // K window [h0-8, h0+24) covers band + halo
        const int r0 = k0 + khalf * 2;
        const int r1 = r0 + 1;
        const int r0c = iclamp(r0, 0, kD - 1);
        const int r1c = iclamp(r1, 0, kD - 1);

        v2f B;
        B.x = in[r0c * kD + w0 + n];
        B.y = in[r1c * kD + w0 + n];

        v2f A;
        A.x = (r0 >= 0 && r0 < kD && iabs_(r0 - hm) <= 5) ? 1.0f : 0.0f;
        A.y = (r1 >= 0 && r1 < kD && iabs_(r1 - hm) <= 5) ? 1.0f : 0.0f;

        acc = __builtin_amdgcn_wmma_f32_16x16x4_f32(
            /*neg_a=*/false, A, /*neg_b=*/false, B,
            /*c_mod=*/(short)0, acc, /*reuse_a=*/false, /*reuse_b=*/false);
    }

#pragma unroll
    for (int v = 0; v < 8; ++v) {
        const int row = h0 + v + khalf * 8;
        out[row * kD + w0 + n] = acc[v];
    }
}

// ---------------------------------------------------------------------------
// Pass 3: sliding box-sum along D (stride kPlane) + per-voxel SSIM + partial
// mean. One thread per (b,h,w) column; enter/leave reads are L2-resident.
// ---------------------------------------------------------------------------
__global__ __launch_bounds__(256) void pass3_d(const float* __restrict__ S2,
                                               float* __restrict__ ssim_partial) {
    __shared__ float red[256];

    const int g  = blockIdx.x * 256 + threadIdx.x;   // 0 .. kLines-1 (162*256 == 41472)
    const int b  = g / kPlane;
    const int hw = g % kPlane;

    const float* __restrict__ c0 = S2 + 0 * kN + b * kVol + hw;
    const float* __restrict__ c1 = S2 + 1 * kN + b * kVol + hw;
    const float* __restrict__ c2 = S2 + 2 * kN + b * kVol + hw;
    const float* __restrict__ c3 = S2 + 3 * kN + b * kVol + hw;
    const float* __restrict__ c4 = S2 + 4 * kN + b * kVol + hw;

    // Initial window for d=0: d' in [0,5]
    float r0 = 0.f, r1 = 0.f, r2 = 0.f, r3 = 0.f, r4 = 0.f;
#pragma unroll
    for (int dd = 0; dd < 6; ++dd) {
        const int o = dd * kPlane;
        r0 += c0[o]; r1 += c1[o]; r2 += c2[o]; r3 += c3[o]; r4 += c4[o];
    }

    float acc = 0.0f;
    for (int d = 0; d < kD; ++d) {
        const float mu_p = r0 * kInvWin3;
        const float mu_t = r1 * kInvWin3;
        const float ep2  = r2 * kInvWin3;
        const float et2  = r3 * kInvWin3;
        const float ept  = r4 * kInvWin3;

        const float mu_p2 = mu_p * mu_p;
        const float mu_t2 = mu_t * mu_t;
        const float mu_pt = mu_p * mu_t;
        const float sp2   = ep2 - mu_p2;
        const float st2   = et2 - mu_t2;
        const float spt   = ept - mu_pt;

        const float num = (2.0f * mu_pt + kC1) * (2.0f * spt + kC2);
        const float den = (mu_p2 + mu_t2 + kC1) * (sp2 + st2 + kC2);
        acc += num / den;

        // Slide window: for d+1 the window is [d-4, d+6]
        if (d + 6 < kD) {
            const int o = (d + 6) * kPlane;
            r0 += c0[o]; r1 += c1[o]; r2 += c2[o]; r3 += c3[o]; r4 += c4[o];
        }
        if (d - 5 >= 0) {
            const int o = (d - 5) * kPlane;
            r0 -= c0[o]; r1 -= c1[o]; r2 -= c2[o]; r3 -= c3[o]; r4 -= c4[o];
        }
    }

    red[threadIdx.x] = acc;
    __syncthreads();
    for (int s = 128; s > 0; s >>= 1) {
        if (threadIdx.x < s) red[threadIdx.x] += red[threadIdx.x + s];
        __syncthreads();
    }
    if (threadIdx.x == 0) ssim_partial[blockIdx.x] = red[0];
}

// ---------------------------------------------------------------------------
// Finalize: deterministic reduction of both partial arrays; writes
// (total_loss, l1_loss, ssim_loss).
// ---------------------------------------------------------------------------
__global__ __launch_bounds__(256) void finalize(const float* __restrict__ l1_partial,
                                                const float* __restrict__ ssim_partial,
                                                float* __restrict__ out) {
    __shared__ float red[256];
    __shared__ float l1_total_sh;
    const int tid = threadIdx.x;

    float s = 0.0f;
    for (int i = tid; i < kLines; i += 256) s += l1_partial[i];
    red[tid] = s;
    __syncthreads();
    for (int st = 128; st > 0; st >>= 1) {
        if (tid < st) red[tid] += red[tid + st];
        __syncthreads();
    }
    if (tid == 0) l1_total_sh = red[0];
    __syncthreads();

    float t = 0.0f;
    for (int i = tid; i < 162; i += 256) t += ssim_partial[i];
    red[tid] = t;
    __syncthreads();
    for (int st = 128; st > 0; st >>= 1) {
        if (tid < st) red[tid] += red[tid + st];
        __syncthreads();
    }

    if (tid == 0) {
        const float inv_n     = 1.0f / (float)kN;
        const float l1_loss   = l1_total_sh * inv_n;
        const float ssim_mean = red[0] * inv_n;
        const float ssim_loss = 1.0f - ssim_mean;
        const float total     = 1.0f * l1_loss + 0.5f * ssim_loss;
        out[0] = total;
        out[1] = l1_loss;
        out[2] = ssim_loss;
    }
}

// ---------------------------------------------------------------------------
// Launch. Workspace layout (floats):
//   S1: 5*kN | S2: 5*kN | l1 partials: kLines | ssim partials: 162
// ~239 MB total.
// ---------------------------------------------------------------------------
extern "C" void kernel_launch(void* const* d_in, const int* in_sizes, int n_in,
                              void* d_out, int out_size, void* d_ws, size_t ws_size,
                              hipStream_t stream) {
    const float* pred   = (const float*)d_in[0];
    const float* target = (const float*)d_in[1];
    float*       out    = (float*)d_out;

    float* S1  = (float*)d_ws;
    float* S2  = S1 + (size_t)5 * kN;
    float* l1p = S2 + (size_t)5 * kN;
    float* sp  = l1p + kLines;

    pass1_w<<<kLines, 256, 0, stream>>>(pred, target, S1, l1p);
    pass2_h<<<dim3(81, 288, 5), 32, 0, stream>>>(S1, S2);
    pass3_d<<<kLines / 256, 256, 0, stream>>>(S2, sp);
    finalize<<<1, 256, 0, stream>>>(l1p, sp, out);
}